// ConceptBank_37306085933420
// MI455X (gfx1250) — compile-verified
//
#include <hip/hip_runtime.h>

// ConceptBank: n-gram (n=2..5) rolling-hash embedding bag + mean pool + L2 norm.
// Hash identity: h(i,n) = sum_{j=0..n-1} (x[i-j]+1)*257^j  < 2^41 << 2^61-1,
// so the Mersenne reduction is a no-op and every position is independent.
// Memory-gather bound: ~256 MB of random 1KB row reads -> ~11us at 23.3 TB/s.

#define T_LEN   2048
#define D_DIM   256
#define CHUNK   128
#define SPLITS  (T_LEN / CHUNK)   // 16
#define NGRAM_COUNT (4 * T_LEN - 10)  // 8182 valid (n,i) pairs per sequence

typedef __attribute__((address_space(1))) int g_int;   // global (AS1)
typedef __attribute__((address_space(3))) int l_int;   // LDS (AS3)

__global__ __launch_bounds__(256) void cb_gather_kernel(
    const int* __restrict__ xb, const float* __restrict__ emb,
    float* __restrict__ partial, int vocab)
{
    const int b    = blockIdx.x;   // batch row
    const int s    = blockIdx.y;   // split along T
    const int t    = threadIdx.x;  // 0..255
    const int pos0 = s * CHUNK;

    __shared__ int          sh_bytes[CHUNK + 4];
    __shared__ unsigned int sh_ids[CHUNK * 4];
    __shared__ float        sh_red[4 * D_DIM];

    // ---- stage byte window [pos0-4, pos0+CHUNK) into LDS (async DMA path) ----
#if __has_builtin(__builtin_amdgcn_global_load_async_to_lds_b32)
    if (t < CHUNK + 4) {
        const int gp = pos0 - 4 + t;
        if (gp >= 0) {
            __builtin_amdgcn_global_load_async_to_lds_b32(
                (g_int*)(xb + (size_t)b * T_LEN + gp),
                (l_int*)&sh_bytes[t], /*offset=*/0, /*cpol=*/0);
        } else {
            sh_bytes[t] = 0;
        }
    }
#if __has_builtin(__builtin_amdgcn_s_wait_asynccnt)
    __builtin_amdgcn_s_wait_asynccnt(0);
#else
    asm volatile("s_wait_asynccnt 0x0" ::: "memory");
#endif
#else
    if (t < CHUNK + 4) {
        const int gp = pos0 - 4 + t;
        sh_bytes[t] = (gp >= 0) ? xb[(size_t)b * T_LEN + gp] : 0;
    }
#endif
    __syncthreads();

    // ---- direct hash -> embedding ids (no scan; raw sum < 2^41 < 2^61-1) ----
    if (t < CHUNK) {
        const int i = pos0 + t;  // global position in sequence
        unsigned long long acc =
            (unsigned long long)(sh_bytes[t + 4] + 1);                    // j=0
        acc += (unsigned long long)(sh_bytes[t + 3] + 1) * 257ull;        // n=2
        sh_ids[t * 4 + 0] = (i >= 1) ? (unsigned)(acc % (unsigned long long)vocab)
                                     : 0xFFFFFFFFu;
        acc += (unsigned long long)(sh_bytes[t + 2] + 1) * 66049ull;      // n=3
        sh_ids[t * 4 + 1] = (i >= 2) ? (unsigned)(acc % (unsigned long long)vocab)
                                     : 0xFFFFFFFFu;
        acc += (unsigned long long)(sh_bytes[t + 1] + 1) * 16974593ull;   // n=4
        sh_ids[t * 4 + 2] = (i >= 3) ? (unsigned)(acc % (unsigned long long)vocab)
                                     : 0xFFFFFFFFu;
        acc += (unsigned long long)(sh_bytes[t + 0] + 1) * 4362470401ull; // n=5
        sh_ids[t * 4 + 3] = (i >= 4) ? (unsigned)(acc % (unsigned long long)vocab)
                                     : 0xFFFFFFFFu;
    }
    __syncthreads();

    // ---- gather + accumulate: 4 row-slots x 64 lanes, float4 per lane ----
    // rs is constant across each 32-lane wave, so the id is wave-uniform:
    // readfirstlane makes it an SGPR -> scalar validity branch + saddr-form
    // global_load_b128 with only the per-lane 16B offset in a VGPR.
    const int rs = t >> 6;   // which of the 4 ids of a position (wave-uniform)
    const int l  = t & 63;   // 16B lane within the 1KB row
    float4 acc4 = make_float4(0.f, 0.f, 0.f, 0.f);
#pragma unroll 8
    for (int p = 0; p < CHUNK; ++p) {
        const unsigned int id =
            (unsigned int)__builtin_amdgcn_readfirstlane((int)sh_ids[p * 4 + rs]);
        if (id != 0xFFFFFFFFu) {  // scalar branch (uniform)
            const float* row = emb + ((size_t)id << 8);  // id * D_DIM, SGPR base
            const float4 v = *(const float4*)(row + (l << 2));
            acc4.x += v.x; acc4.y += v.y; acc4.z += v.z; acc4.w += v.w;
        }
    }

    // ---- cross-slot reduction in LDS, write per-split partial sum ----
    *(float4*)&sh_red[rs * D_DIM + (l << 2)] = acc4;
    __syncthreads();
    const float sum = sh_red[t] + sh_red[D_DIM + t] +
                      sh_red[2 * D_DIM + t] + sh_red[3 * D_DIM + t];
    partial[((size_t)b * SPLITS + s) * D_DIM + t] = sum;
}

__global__ __launch_bounds__(256) void cb_final_kernel(
    const float* __restrict__ partial, float* __restrict__ out)
{
    const int b = blockIdx.x;
    const int t = threadIdx.x;  // dim 0..255

    float sum = 0.f;
#pragma unroll
    for (int s = 0; s < SPLITS; ++s)
        sum += partial[((size_t)b * SPLITS + s) * D_DIM + t];

    const float mean = sum * (1.0f / (float)NGRAM_COUNT);

    __shared__ float red[D_DIM];
    red[t] = mean * mean;
    __syncthreads();
    for (int off = D_DIM / 2; off > 0; off >>= 1) {
        if (t < off) red[t] += red[t + off];
        __syncthreads();
    }
    const float norm = sqrtf(red[0]);
    out[(size_t)b * D_DIM + t] = mean / fmaxf(norm, 1e-12f);
}

extern "C" void kernel_launch(void* const* d_in, const int* in_sizes, int n_in,
                              void* d_out, int out_size, void* d_ws, size_t ws_size,
                              hipStream_t stream) {
    const int*   xb  = (const int*)d_in[0];     // [B, 2048] int32 byte values
    const float* emb = (const float*)d_in[1];   // [vocab, 256] f32
    float*       out = (float*)d_out;           // [B, 256] f32
    float*       partial = (float*)d_ws;        // [B, SPLITS, 256] f32 scratch

    const int B     = in_sizes[0] / T_LEN;      // 32
    const int vocab = in_sizes[1] / D_DIM;      // 1,000,000

    dim3 grid1(B, SPLITS);
    cb_gather_kernel<<<grid1, 256, 0, stream>>>(xb, emb, partial, vocab);
    cb_final_kernel<<<B, 256, 0, stream>>>(partial, out);
}